// TetrahedralConvolution_73547019976726
// MI455X (gfx1250) — compile-verified
//
#include <hip/hip_runtime.h>

typedef __attribute__((ext_vector_type(2))) float v2f;
typedef __attribute__((ext_vector_type(8))) float v8f;

#define B_     4
#define C_IN_  32
#define C_OUT_ 64
#define P_     100000
#define K_     8
#define TILE_P 64           // positions per block -> 16 N-columns per group g
#define PADP   65           // padded row length for agg_s (bank-conflict free)
#define NTHR   512          // 16 wave32 waves: wave w -> (b = w>>2, g = w&3)

__global__ __launch_bounds__(NTHR)
void tet_conv_fused_kernel(const float* __restrict__ x,
                           const float* __restrict__ weight,
                           const float* __restrict__ bias,
                           const float* __restrict__ geo,
                           const int*   __restrict__ nbr,
                           float* __restrict__ out)
{
    __shared__ float att_s[TILE_P][K_];
    __shared__ int   nbr_s[TILE_P][K_];
    __shared__ float agg_s[B_][C_IN_][PADP];   // [batch][channel][local pos]

    const int tid = threadIdx.x;
    const int p0  = blockIdx.x * TILE_P;

    // ---- Phase 1a: per-position softmax over 8 neighbor logits (threads 0..63)
    if (tid < TILE_P) {
        const int p = p0 + tid;
        if (p < P_) {
            int   n[K_];
            float gv[K_];
            float m = -3.402823466e38f;
#pragma unroll
            for (int k = 0; k < K_; ++k) {
                n[k]  = nbr[p * K_ + k];
                gv[k] = geo[n[k]];
                m = fmaxf(m, gv[k]);
            }
            float s = 0.f;
#pragma unroll
            for (int k = 0; k < K_; ++k) { gv[k] = __expf(gv[k] - m); s += gv[k]; }
            const float inv = 1.f / s;
#pragma unroll
            for (int k = 0; k < K_; ++k) { att_s[tid][k] = gv[k] * inv; nbr_s[tid][k] = n[k]; }
        } else {
            // zero-fill tail so WMMA columns compute 0 (EXEC must stay all-ones)
#pragma unroll
            for (int k = 0; k < K_; ++k) { att_s[tid][k] = 0.f; nbr_s[tid][k] = 0; }
        }
    }
    __syncthreads();

    // ---- Phase 1b: gather + attention-weighted aggregate for ALL 4 batches.
    // lane == channel  ->  nbr_s/att_s reads are wave-broadcast (no conflicts)
    {
        const int c = tid & 31;                       // lane
        for (int it = 0; it < (B_ * C_IN_ * TILE_P) / NTHR; ++it) {
            const int rest = (it * NTHR + tid) >> 5;  // 0..255
            const int pl   = rest & 63;
            const int bb   = rest >> 6;
            const float* __restrict__ xr =
                x + ((size_t)bb * C_IN_ + c) * P_;
            float acc = 0.f;
#pragma unroll
            for (int k = 0; k < K_; ++k)
                acc += xr[nbr_s[pl][k]] * att_s[pl][k];
            agg_s[bb][c][pl] = acc;
        }
    }
    __syncthreads();

    // ---- Phase 2: WMMA f32 16x16x4.  wave (bb,g) handles columns p = p0 + 4q + g.
    // A (16x4):  lane(0..15)=M row; VGPR0={K0 | K2}, VGPR1={K1 | K3}
    // B (4x16):  lane&15 = N col;   VGPR0={K0 | K2}, VGPR1={K1 | K3}
    // D (16x16): VGPR r: lanes 0-15 -> M=r, lanes 16-31 -> M=r+8
    const int wave = tid >> 5;           // 0..15
    const int lane = tid & 31;
    const int bb   = wave >> 2;
    const int g    = wave & 3;
    const int col  = lane & 15;          // N index / A's M row
    const int kh   = lane >> 4;          // 0 -> K{0,1}, 1 -> K{2,3}
    const int pl   = 4 * col + g;        // local position of this N column
    const int p    = p0 + pl;

    // Hoist B operands: identical for all 4 M-tiles.
    v2f bmat[C_IN_ / 4];
#pragma unroll
    for (int ks = 0; ks < C_IN_ / 4; ++ks) {
        const int k0 = ks * 4 + 2 * kh;
        bmat[ks].x = agg_s[bb][k0    ][pl];
        bmat[ks].y = agg_s[bb][k0 + 1][pl];
    }

#pragma unroll
    for (int mt = 0; mt < 4; ++mt) {     // M = 64 -> 4 tiles of 16
        const int o0 = mt * 16;
        v8f acc = {};
#pragma unroll
        for (int ks = 0; ks < C_IN_ / 4; ++ks) {   // K = 32 -> 8 steps of 4
            const int k0 = ks * 4 + 2 * kh;
            v2f a;
            a.x = weight[((o0 + col) * C_IN_ + k0    ) * 4 + g];
            a.y = weight[((o0 + col) * C_IN_ + k0 + 1) * 4 + g];
            acc = __builtin_amdgcn_wmma_f32_16x16x4_f32(
                false, a, false, bmat[ks], (short)0, acc, false, false);
        }
        if (p < P_) {
#pragma unroll
            for (int r = 0; r < 8; ++r) {
                const int o = o0 + r + 8 * kh;
                out[((size_t)bb * C_OUT_ + o) * P_ + p] = acc[r] + bias[o];
            }
        }
    }
}

extern "C" void kernel_launch(void* const* d_in, const int* in_sizes, int n_in,
                              void* d_out, int out_size, void* d_ws, size_t ws_size,
                              hipStream_t stream) {
    (void)in_sizes; (void)n_in; (void)out_size; (void)d_ws; (void)ws_size;
    const float* x      = (const float*)d_in[0];
    const float* weight = (const float*)d_in[1];
    const float* bias   = (const float*)d_in[2];
    const float* geo    = (const float*)d_in[3];
    const int*   nbr    = (const int*)  d_in[4];
    float* out = (float*)d_out;

    dim3 grid((P_ + TILE_P - 1) / TILE_P, 1);
    tet_conv_fused_kernel<<<grid, NTHR, 0, stream>>>(x, weight, bias, geo, nbr, out);
}